// LinearSelfAttention_13048110645546
// MI455X (gfx1250) — compile-verified
//
#include <hip/hip_runtime.h>

// ---------------------------------------------------------------------------
// Chunked causal linear attention for gfx1250 (MI455X), full f32 via
// V_WMMA_F32_16X16X4_F32.  All WMMA operands stored in LDS so that both A and
// B fragments are contiguous 8-byte pairs (single ds_load_b64, no repacking).
//
// Shapes: B=4, T=2048, nh=8, N=64, D=128.  Chunk C=64, NC=32 chunks.
//   P[b,h,c]  = Q_c^T V_c                         (kernel 1, WMMA)
//   S[b,h,c]  = state + sum_{c'<c} P_{c'}         (kernel 2, prefix)
//   new_state = state + sum_c P_c                 (kernel 2)
//   out_c     = Q_c S_c + tril(Q_c Q_c^T,-1) V_c  (kernel 3, WMMA)
// ---------------------------------------------------------------------------

typedef __attribute__((ext_vector_type(2))) float v2f;
typedef __attribute__((ext_vector_type(8))) float v8f;

#define BB 4
#define TT 2048
#define NH 8
#define NN 64      // head key/query dim
#define DD 128     // value dim
#define CC 64      // chunk length
#define NCHUNK (TT / CC)   // 32
#define PAD 68     // LDS row stride (dwords); 68 % 64 == 4 -> b64 reads tile all banks

#define WMMA_F32(a, b, c) \
    __builtin_amdgcn_wmma_f32_16x16x4_f32(false, (a), false, (b), (short)0, (c), false, false)

// ---------------------------------------------------------------------------
// Kernel 1: P[b,h,c] = Q_c^T V_c   (N x D = 64 x 128, contraction over time C)
// grid = B*NH*NCHUNK, 256 threads (8 waves); wave -> 4 output tiles.
// ---------------------------------------------------------------------------
__global__ __launch_bounds__(256)
void lsa_chunk_outer_kernel(const float* __restrict__ Q,
                            const float* __restrict__ V,
                            float* __restrict__ P) {
    __shared__ __align__(16) float Qt[NN][PAD];   // [n][t]
    __shared__ __align__(16) float Vt[DD][PAD];   // [d][t]

    const int blk = blockIdx.x;            // (b*NH + h)*NCHUNK + c
    const int c   = blk % NCHUNK;
    const int bh  = blk / NCHUNK;
    const int b   = bh / NH;
    const int h   = bh % NH;
    const int t0  = c * CC;
    const int tid = threadIdx.x;

    // Q chunk: b128 loads (coalesced over n), transposed scatter into LDS
    #pragma unroll
    for (int r = 0; r < 4; ++r) {
        int e = tid + 256 * r;             // 0..1023 float4s
        int t = e >> 4, n4 = (e & 15) * 4;
        const float4 q = *reinterpret_cast<const float4*>(
            &Q[((size_t)(b * TT + t0 + t) * NH + h) * NN + n4]);
        Qt[n4 + 0][t] = q.x; Qt[n4 + 1][t] = q.y;
        Qt[n4 + 2][t] = q.z; Qt[n4 + 3][t] = q.w;
    }
    // V chunk: b128 loads, transposed scatter
    #pragma unroll
    for (int r = 0; r < 8; ++r) {
        int e = tid + 256 * r;             // 0..2047 float4s
        int t = e >> 5, d4 = (e & 31) * 4;
        const float4 v = *reinterpret_cast<const float4*>(
            &V[(size_t)(b * TT + t0 + t) * DD + d4]);
        Vt[d4 + 0][t] = v.x; Vt[d4 + 1][t] = v.y;
        Vt[d4 + 2][t] = v.z; Vt[d4 + 3][t] = v.w;
    }
    __syncthreads();

    const int wave = tid >> 5;
    const int lane = tid & 31;
    const int l15  = lane & 15;
    const int hi   = lane >> 4;
    const int mBase = (wave & 3) * 16;     // row tile over N
    const int jBase = (wave >> 2) * 4;     // col tile group over D

    const float* aRow = &Qt[mBase + l15][0];
    const float* bR0  = &Vt[(jBase + 0) * 16 + l15][0];
    const float* bR1  = &Vt[(jBase + 1) * 16 + l15][0];
    const float* bR2  = &Vt[(jBase + 2) * 16 + l15][0];
    const float* bR3  = &Vt[(jBase + 3) * 16 + l15][0];

    v8f acc0 = {}, acc1 = {}, acc2 = {}, acc3 = {};
    for (int k = 0; k < CC; k += 4) {
        const int kk = k + 2 * hi;
        v2f a  = *(const v2f*)(aRow + kk);
        v2f b0 = *(const v2f*)(bR0 + kk);
        v2f b1 = *(const v2f*)(bR1 + kk);
        v2f b2 = *(const v2f*)(bR2 + kk);
        v2f b3 = *(const v2f*)(bR3 + kk);
        acc0 = WMMA_F32(a, b0, acc0);
        acc1 = WMMA_F32(a, b1, acc1);
        acc2 = WMMA_F32(a, b2, acc2);
        acc3 = WMMA_F32(a, b3, acc3);
    }

    float* Pc = P + ((size_t)bh * NCHUNK + c) * (NN * DD);
    v8f accs[4] = {acc0, acc1, acc2, acc3};
    #pragma unroll
    for (int j = 0; j < 4; ++j) {
        int col = (jBase + j) * 16 + l15;
        #pragma unroll
        for (int v = 0; v < 8; ++v) {
            int row = mBase + v + 8 * hi;
            Pc[(size_t)row * DD + col] = accs[j][v];
        }
    }
}

// ---------------------------------------------------------------------------
// Kernel 2: exclusive prefix over chunks + final state (vectorized streaming).
// ---------------------------------------------------------------------------
__global__ __launch_bounds__(256)
void lsa_prefix_kernel(const float4* __restrict__ P4,
                       const float4* __restrict__ state4,
                       float4* __restrict__ S4,
                       float4* __restrict__ new_state4) {
    const int bh  = blockIdx.x;
    const int tid = threadIdx.x;
    const int M4  = (NN * DD) / 4;              // 2048 float4s per matrix
    #pragma unroll
    for (int e = tid; e < M4; e += 256) {
        float4 acc = state4[(size_t)bh * M4 + e];
        size_t pc = (size_t)bh * NCHUNK * M4 + e;
        for (int c = 0; c < NCHUNK; ++c) {
            S4[pc] = acc;                        // exclusive prefix
            float4 p = P4[pc];
            acc.x += p.x; acc.y += p.y; acc.z += p.z; acc.w += p.w;
            pc += M4;
        }
        new_state4[(size_t)bh * M4 + e] = acc;
    }
}

// ---------------------------------------------------------------------------
// Kernel 3: out_c = Q_c S_c + tril(Q_c Q_c^T, -1) V_c   (C x D = 64 x 128)
// Phase 1: 16 score tiles (64x64), masked, staged through LDS.
// Phase 2: 32 output tiles, two contraction chains, hoisted A fragments.
// ---------------------------------------------------------------------------
__global__ __launch_bounds__(256)
void lsa_output_kernel(const float* __restrict__ Q,
                       const float* __restrict__ V,
                       const float* __restrict__ S,
                       float* __restrict__ out) {
    __shared__ __align__(16) float Qs[CC][PAD];   // [t][n]
    __shared__ __align__(16) float Sc[CC][PAD];   // masked scores [t][u]
    __shared__ __align__(16) float Vt[DD][PAD];   // [d][t]
    __shared__ __align__(16) float St[DD][PAD];   // state chunk [d][n]

    const int blk = blockIdx.x;
    const int c   = blk % NCHUNK;
    const int bh  = blk / NCHUNK;
    const int b   = bh / NH;
    const int h   = bh % NH;
    const int t0  = c * CC;
    const int tid = threadIdx.x;

    // Q chunk -> Qs[t][n] (b128 load, b128 LDS store)
    #pragma unroll
    for (int r = 0; r < 4; ++r) {
        int e = tid + 256 * r;
        int t = e >> 4, n4 = (e & 15) * 4;
        *reinterpret_cast<float4*>(&Qs[t][n4]) = *reinterpret_cast<const float4*>(
            &Q[((size_t)(b * TT + t0 + t) * NH + h) * NN + n4]);
    }
    // V chunk -> Vt[d][t] (transposed)
    #pragma unroll
    for (int r = 0; r < 8; ++r) {
        int e = tid + 256 * r;
        int t = e >> 5, d4 = (e & 31) * 4;
        const float4 v = *reinterpret_cast<const float4*>(
            &V[(size_t)(b * TT + t0 + t) * DD + d4]);
        Vt[d4 + 0][t] = v.x; Vt[d4 + 1][t] = v.y;
        Vt[d4 + 2][t] = v.z; Vt[d4 + 3][t] = v.w;
    }
    // S chunk [n][d] -> St[d][n] (transposed)
    const float* Schunk = S + ((size_t)bh * NCHUNK + c) * (NN * DD);
    #pragma unroll
    for (int r = 0; r < 8; ++r) {
        int e = tid + 256 * r;
        int n = e >> 5, d4 = (e & 31) * 4;
        const float4 s = *reinterpret_cast<const float4*>(&Schunk[(size_t)n * DD + d4]);
        St[d4 + 0][n] = s.x; St[d4 + 1][n] = s.y;
        St[d4 + 2][n] = s.z; St[d4 + 3][n] = s.w;
    }
    __syncthreads();

    const int wave = tid >> 5;
    const int lane = tid & 31;
    const int l15  = lane & 15;
    const int hi   = lane >> 4;

    // ---- Phase 1: scores = tril(Q Q^T, -1): 16 tiles, 2 per wave -----------
    {
        const int i  = wave & 3;             // row tile (t)
        const int j0 = (wave >> 2) * 2;      // two col tiles (u)
        const float* aRow  = &Qs[i * 16 + l15][0];
        const float* bRow0 = &Qs[(j0 + 0) * 16 + l15][0];
        const float* bRow1 = &Qs[(j0 + 1) * 16 + l15][0];
        v8f s0 = {}, s1 = {};
        for (int k = 0; k < NN; k += 4) {
            const int kk = k + 2 * hi;
            v2f a  = *(const v2f*)(aRow + kk);
            v2f b0 = *(const v2f*)(bRow0 + kk);
            v2f b1 = *(const v2f*)(bRow1 + kk);
            s0 = WMMA_F32(a, b0, s0);
            s1 = WMMA_F32(a, b1, s1);
        }
        // strict-causal mask + stage (C-layout -> row-major LDS)
        v8f sc[2] = {s0, s1};
        #pragma unroll
        for (int jj = 0; jj < 2; ++jj) {
            const int colIdx = (j0 + jj) * 16 + l15;
            #pragma unroll
            for (int v = 0; v < 8; ++v) {
                int rowIdx = i * 16 + v + 8 * hi;
                Sc[rowIdx][colIdx] = (rowIdx > colIdx) ? sc[jj][v] : 0.0f;
            }
        }
    }
    __syncthreads();

    // ---- Phase 2: out = Q @ S_c + scores @ V: 32 tiles, 4 per wave ---------
    {
        const int i     = wave & 3;          // row tile over time
        const int jBase = (wave >> 2) * 4;   // col tiles over D
        const float* aQ  = &Qs[i * 16 + l15][0];
        const float* aSc = &Sc[i * 16 + l15][0];
        const float* bS0 = &St[(jBase + 0) * 16 + l15][0];
        const float* bS1 = &St[(jBase + 1) * 16 + l15][0];
        const float* bS2 = &St[(jBase + 2) * 16 + l15][0];
        const float* bS3 = &St[(jBase + 3) * 16 + l15][0];
        const float* bV0 = &Vt[(jBase + 0) * 16 + l15][0];
        const float* bV1 = &Vt[(jBase + 1) * 16 + l15][0];
        const float* bV2 = &Vt[(jBase + 2) * 16 + l15][0];
        const float* bV3 = &Vt[(jBase + 3) * 16 + l15][0];

        v8f acc0 = {}, acc1 = {}, acc2 = {}, acc3 = {};
        // Chain 1: contraction over N (Q @ S_c)
        for (int k = 0; k < NN; k += 4) {
            const int kk = k + 2 * hi;
            v2f a  = *(const v2f*)(aQ + kk);
            v2f b0 = *(const v2f*)(bS0 + kk);
            v2f b1 = *(const v2f*)(bS1 + kk);
            v2f b2 = *(const v2f*)(bS2 + kk);
            v2f b3 = *(const v2f*)(bS3 + kk);
            acc0 = WMMA_F32(a, b0, acc0);
            acc1 = WMMA_F32(a, b1, acc1);
            acc2 = WMMA_F32(a, b2, acc2);
            acc3 = WMMA_F32(a, b3, acc3);
        }
        // Chain 2: contraction over U (scores @ V)
        for (int k = 0; k < CC; k += 4) {
            const int kk = k + 2 * hi;
            v2f a  = *(const v2f*)(aSc + kk);
            v2f b0 = *(const v2f*)(bV0 + kk);
            v2f b1 = *(const v2f*)(bV1 + kk);
            v2f b2 = *(const v2f*)(bV2 + kk);
            v2f b3 = *(const v2f*)(bV3 + kk);
            acc0 = WMMA_F32(a, b0, acc0);
            acc1 = WMMA_F32(a, b1, acc1);
            acc2 = WMMA_F32(a, b2, acc2);
            acc3 = WMMA_F32(a, b3, acc3);
        }
        // Store: out[b, t0+row, h, col]
        v8f accs[4] = {acc0, acc1, acc2, acc3};
        #pragma unroll
        for (int j = 0; j < 4; ++j) {
            int col = (jBase + j) * 16 + l15;
            #pragma unroll
            for (int v = 0; v < 8; ++v) {
                int row = i * 16 + v + 8 * hi;
                out[((size_t)(b * TT + t0 + row) * NH + h) * DD + col] = accs[j][v];
            }
        }
    }
}

// ---------------------------------------------------------------------------
extern "C" void kernel_launch(void* const* d_in, const int* in_sizes, int n_in,
                              void* d_out, int out_size, void* d_ws, size_t ws_size,
                              hipStream_t stream) {
    (void)in_sizes; (void)n_in; (void)out_size; (void)ws_size;
    const float* Q     = (const float*)d_in[0];   // [B,T,nh,N]
    const float* V     = (const float*)d_in[1];   // [B,T,D]
    const float* state = (const float*)d_in[2];   // [B,nh,N,D]
    // d_in[3] = return_new_state (always 1 in this harness)

    float* out       = (float*)d_out;                               // [B,T,nh,D]
    float* new_state = out + (size_t)BB * TT * NH * DD;             // [B,nh,N,D]

    const size_t chunkMat = (size_t)BB * NH * NCHUNK * NN * DD;     // 8.39M floats
    float* P = (float*)d_ws;            // per-chunk outer products
    float* S = P + chunkMat;            // exclusive prefix states

    lsa_chunk_outer_kernel<<<BB * NH * NCHUNK, 256, 0, stream>>>(Q, V, P);
    lsa_prefix_kernel<<<BB * NH, 256, 0, stream>>>(
        (const float4*)P, (const float4*)state, (float4*)S, (float4*)new_state);
    lsa_output_kernel<<<BB * NH * NCHUNK, 256, 0, stream>>>(Q, V, S, out);
}